// MLP321_55259049230985
// MI455X (gfx1250) — compile-verified
//
#include <hip/hip_runtime.h>
#include <math.h>

typedef __attribute__((ext_vector_type(2))) float v2f;
typedef __attribute__((ext_vector_type(8))) float v8f;

// wave32 gather across lanes via DS_BPERMUTE_B32 (byte address = lane*4)
__device__ __forceinline__ float lane_gather(float v, int srcLane) {
    union { float f; int i; } u; u.f = v;
    int r = __builtin_amdgcn_ds_bpermute(srcLane << 2, u.i);
    union { int i; float f; } o; o.i = r;
    return o.f;
}

// Main kernel: 256 threads = 8 waves; each wave handles 32 consecutive rows
// as two 16-column WMMA tiles. Layers 1 and 2 computed as
//   H^T = [W | b] (A, 16x4 zero-padded) x [features ; 1] (B, 4x16)
// with V_WMMA_F32_16X16X4_F32. Requires gridDim.x*256 rows exactly (EXEC all-1s).
extern "C" __global__ __launch_bounds__(256)
void mlp321_wmma_main(const float* __restrict__ x,
                      const float* __restrict__ W1, const float* __restrict__ B1,
                      const float* __restrict__ W2, const float* __restrict__ B2,
                      const float* __restrict__ W3, const float* __restrict__ B3,
                      float* __restrict__ outp, float* __restrict__ h1p,
                      float* __restrict__ h2p)
{
    const int  lane = threadIdx.x & 31;
    const int  wv   = threadIdx.x >> 5;
    const int  m    = lane & 15;          // row-in-A / column-in-tile index
    const bool hi   = lane >= 16;         // upper K half (K=2,3)

    const size_t r = ((size_t)blockIdx.x * 8 + wv) * 32 + lane;

    // Coalesced, non-temporal b64 load: lane n holds features of row base+n.
    v2f xv = __builtin_nontemporal_load((const v2f*)(x + 2 * r));

    // Uniform scalar weight loads (20 floats, cache-resident).
    const float w100 = W1[0], w101 = W1[1];
    const float w110 = W1[2], w111 = W1[3];
    const float w120 = W1[4], w121 = W1[5];
    const float b10 = B1[0], b11 = B1[1], b12 = B1[2];
    const float w200 = W2[0], w201 = W2[1], w202 = W2[2];
    const float w210 = W2[3], w211 = W2[4], w212 = W2[5];
    const float b20 = B2[0], b21 = B2[1];
    const float w300 = W3[0], w301 = W3[1], b30 = B3[0];

    // ---- Layer 1 A matrix: rows m = hidden unit, K = {x0, x1, bias, 0} ----
    // A 16x4 layout: v0 = K0 (lanes 0-15) | K2 (lanes 16-31); v1 = K1 | K3.
    v2f A1;
    {
        float k0 = (m == 0) ? w100 : (m == 1) ? w110 : (m == 2) ? w120 : 0.f;
        float k1 = (m == 0) ? w101 : (m == 1) ? w111 : (m == 2) ? w121 : 0.f;
        float k2 = (m == 0) ? b10  : (m == 1) ? b11  : (m == 2) ? b12  : 0.f;
        A1.x = hi ? k2 : k0;
        A1.y = hi ? 0.f : k1;
    }

    // ---- Layer 1 B matrices (4x16): columns = batch rows of this tile ----
    // B layout: v0 = K0 (lanes 0-15) | K2 (lanes 16-31); v1 = K1 | K3.
    float x0a = lane_gather(xv.x, m);        // tile A: rows base..base+15
    float x1a = lane_gather(xv.y, m);
    float x0b = lane_gather(xv.x, m + 16);   // tile B: rows base+16..base+31
    float x1b = lane_gather(xv.y, m + 16);
    v2f Ba; Ba.x = hi ? 1.f : x0a; Ba.y = hi ? 0.f : x1a;  // K2 = ones (bias)
    v2f Bb; Bb.x = hi ? 1.f : x0b; Bb.y = hi ? 0.f : x1b;

    v8f cz = {0.f, 0.f, 0.f, 0.f, 0.f, 0.f, 0.f, 0.f};
    v8f d1a = __builtin_amdgcn_wmma_f32_16x16x4_f32(false, A1, false, Ba,
                                                    (short)0, cz, false, false);
    v8f d1b = __builtin_amdgcn_wmma_f32_16x16x4_f32(false, A1, false, Bb,
                                                    (short)0, cz, false, false);

    // D layout: vgpr v, lanes 0-15 hold (m=v, n=lane). Hidden units 0..2 valid.
    float h1a0 = fmaxf(d1a[0], 0.f), h1a1 = fmaxf(d1a[1], 0.f), h1a2 = fmaxf(d1a[2], 0.f);
    float h1b0 = fmaxf(d1b[0], 0.f), h1b1 = fmaxf(d1b[1], 0.f), h1b2 = fmaxf(d1b[2], 0.f);

    // Merge tiles so lane n owns row base+n, then store h1 (contiguous bytes).
    float u0 = lane_gather(h1b0, m), u1 = lane_gather(h1b1, m), u2 = lane_gather(h1b2, m);
    float H10 = hi ? u0 : h1a0;
    float H11 = hi ? u1 : h1a1;
    float H12 = hi ? u2 : h1a2;
    __builtin_nontemporal_store(H10, h1p + 3 * r + 0);
    __builtin_nontemporal_store(H11, h1p + 3 * r + 1);
    __builtin_nontemporal_store(H12, h1p + 3 * r + 2);

    // ---- Layer 2 A matrix: rows m = hidden2 unit, K = {h1_0,h1_1,h1_2,bias} ----
    v2f A2;
    {
        float k0 = (m == 0) ? w200 : (m == 1) ? w210 : 0.f;
        float k1 = (m == 0) ? w201 : (m == 1) ? w211 : 0.f;
        float k2 = (m == 0) ? w202 : (m == 1) ? w212 : 0.f;
        float k3 = (m == 0) ? b20  : (m == 1) ? b21  : 0.f;
        A2.x = hi ? k2 : k0;
        A2.y = hi ? k3 : k1;
    }

    // ---- Layer 2 B matrices: K rows {h1[n][0], h1[n][1], h1[n][2], 1} ----
    float g2a = lane_gather(h1a2, m);  // K2 values into lanes 16-31
    float g2b = lane_gather(h1b2, m);
    v2f B2a; B2a.x = hi ? g2a : h1a0; B2a.y = hi ? 1.f : h1a1;
    v2f B2b; B2b.x = hi ? g2b : h1b0; B2b.y = hi ? 1.f : h1b1;

    v8f d2a = __builtin_amdgcn_wmma_f32_16x16x4_f32(false, A2, false, B2a,
                                                    (short)0, cz, false, false);
    v8f d2b = __builtin_amdgcn_wmma_f32_16x16x4_f32(false, A2, false, B2b,
                                                    (short)0, cz, false, false);

    float h2a0 = fmaxf(d2a[0], 0.f), h2a1 = fmaxf(d2a[1], 0.f);
    float h2b0 = fmaxf(d2b[0], 0.f), h2b1 = fmaxf(d2b[1], 0.f);
    float v0 = lane_gather(h2b0, m), v1 = lane_gather(h2b1, m);
    float H20 = hi ? v0 : h2a0;
    float H21 = hi ? v1 : h2a1;
    v2f h2v; h2v.x = H20; h2v.y = H21;            // 8 B contiguous, 8 B aligned
    __builtin_nontemporal_store(h2v, (v2f*)(h2p + 2 * r));

    // ---- Layer 3 (2 -> 1) + sigmoid: plain VALU, full-wave coalesced store ----
    // sigmoid via v_exp_f32 + v_rcp_f32 (avoid the precise v_div_scale sequence)
    float o = fmaf(w300, H20, fmaf(w301, H21, b30));
    float s = __builtin_amdgcn_rcpf(1.f + __expf(-o));
    __builtin_nontemporal_store(s, outp + r);
}

// Scalar tail kernel for rows beyond a multiple of 256 (empty for B = 8388608).
extern "C" __global__ void mlp321_scalar_tail(const float* __restrict__ x,
                                              const float* __restrict__ W1,
                                              const float* __restrict__ B1,
                                              const float* __restrict__ W2,
                                              const float* __restrict__ B2,
                                              const float* __restrict__ W3,
                                              const float* __restrict__ B3,
                                              float* __restrict__ outp,
                                              float* __restrict__ h1p,
                                              float* __restrict__ h2p,
                                              unsigned long long start,
                                              unsigned long long nrows)
{
    unsigned long long i = (unsigned long long)blockIdx.x * blockDim.x + threadIdx.x;
    if (i >= nrows) return;
    unsigned long long r = start + i;

    float x0 = x[2 * r + 0], x1 = x[2 * r + 1];
    float h1[3];
    h1[0] = fmaxf(fmaf(W1[0], x0, fmaf(W1[1], x1, B1[0])), 0.f);
    h1[1] = fmaxf(fmaf(W1[2], x0, fmaf(W1[3], x1, B1[1])), 0.f);
    h1[2] = fmaxf(fmaf(W1[4], x0, fmaf(W1[5], x1, B1[2])), 0.f);
    float h2[2];
    h2[0] = fmaxf(fmaf(W2[0], h1[0], fmaf(W2[1], h1[1], fmaf(W2[2], h1[2], B2[0]))), 0.f);
    h2[1] = fmaxf(fmaf(W2[3], h1[0], fmaf(W2[4], h1[1], fmaf(W2[5], h1[2], B2[1]))), 0.f);
    float o = fmaf(W3[0], h2[0], fmaf(W3[1], h2[1], B3[0]));

    h1p[3 * r + 0] = h1[0];
    h1p[3 * r + 1] = h1[1];
    h1p[3 * r + 2] = h1[2];
    h2p[2 * r + 0] = h2[0];
    h2p[2 * r + 1] = h2[1];
    outp[r] = __builtin_amdgcn_rcpf(1.f + __expf(-o));
}

extern "C" void kernel_launch(void* const* d_in, const int* in_sizes, int n_in,
                              void* d_out, int out_size, void* d_ws, size_t ws_size,
                              hipStream_t stream) {
    (void)n_in; (void)d_ws; (void)ws_size; (void)out_size;

    const float* x  = (const float*)d_in[0];
    const float* W1 = (const float*)d_in[1];
    const float* b1 = (const float*)d_in[2];
    const float* W2 = (const float*)d_in[3];
    const float* b2 = (const float*)d_in[4];
    const float* W3 = (const float*)d_in[5];
    const float* b3 = (const float*)d_in[6];

    const size_t rows = (size_t)in_sizes[0] / 2;

    // d_out = [out (rows) | h1 (3*rows) | h2 (2*rows)] concatenated flat.
    float* outp = (float*)d_out;
    float* h1p  = outp + rows;
    float* h2p  = outp + 4 * rows;

    const size_t mainRows = rows & ~(size_t)255;   // 256 rows per block
    const size_t tailRows = rows - mainRows;

    if (mainRows) {
        dim3 grid((unsigned)(mainRows / 256));
        mlp321_wmma_main<<<grid, 256, 0, stream>>>(x, W1, b1, W2, b2, W3, b3,
                                                   outp, h1p, h2p);
    }
    if (tailRows) {
        dim3 grid((unsigned)((tailRows + 255) / 256));
        mlp321_scalar_tail<<<grid, 256, 0, stream>>>(x, W1, b1, W2, b2, W3, b3,
                                                     outp, h1p, h2p,
                                                     (unsigned long long)mainRows,
                                                     (unsigned long long)tailRows);
    }
}